// GCN_56745107914902
// MI455X (gfx1250) — compile-verified
//
#include <hip/hip_runtime.h>
#include <math.h>

#define NN 100000      // nodes
#define NE 3200000     // edges
#define NG 1024        // graphs
#define NC 10          // classes
#define KTOP 30
#define XCF 97         // concat feature width

typedef float v2f __attribute__((ext_vector_type(2)));
typedef float v8f __attribute__((ext_vector_type(8)));

// ---------------- utility ----------------
__global__ void zero_f(float* p, int n) {
    int i = blockIdx.x * blockDim.x + threadIdx.x;
    if (i < n) p[i] = 0.f;
}
__global__ void zero_i(int* p, int n) {
    int i = blockIdx.x * blockDim.x + threadIdx.x;
    if (i < n) p[i] = 0;
}

// ---------------- degree / normalization ----------------
__global__ void deg_edges(const int* __restrict__ src, const int* __restrict__ dst,
                          float* __restrict__ deg) {
    int e = blockIdx.x * blockDim.x + threadIdx.x;
    if (e >= NE) return;
    int s = src[e], d = dst[e];
    if (s != d) atomicAdd(&deg[d], 1.0f);
}
__global__ void make_dis(float* __restrict__ degdis) {
    int i = blockIdx.x * blockDim.x + threadIdx.x;
    if (i < NN) degdis[i] = rsqrtf(degdis[i] + 1.0f);
}

// ---------------- f32 WMMA GEMM: C[MxN] = A[MxK] * B[KxN] (+bias, relu) ----------
// One wave per 16x16 output tile, K stepped by 4 via V_WMMA_F32_16X16X4_F32.
// A-frag: lane(l<16) comps = A[row][k0+0..1]; lanes 16-31 = A[row][k0+2..3].
// B-frag: lane(l<16) comps = B[k0+0..1][col]; lanes 16-31 = B[k0+2..3][col].
// D: VGPR v -> row rt*16 + v + 8*(lane>=16), col = ct*16 + (lane&15).
__global__ void wmma_mm(const float* __restrict__ A, int lda,
                        const float* __restrict__ B, int ldb,
                        float* __restrict__ C, int ldc,
                        int K, int n_row_tiles, int n_col_tiles,
                        const float* __restrict__ bias, int relu) {
    int wave = (blockIdx.x * blockDim.x + threadIdx.x) >> 5;
    int lane = threadIdx.x & 31;
    int rt = wave / n_col_tiles;
    int ct = wave - rt * n_col_tiles;
    if (rt >= n_row_tiles) return;              // whole-wave uniform branch; EXEC stays all-1
    int m = lane & 15;
    int half = lane >> 4;
    int col = ct * 16 + m;
    const float* Arow = A + (size_t)(rt * 16 + m) * lda + 2 * half;
    const float* Bcol = B + (size_t)(2 * half) * ldb + col;
    v8f acc = {0.f, 0.f, 0.f, 0.f, 0.f, 0.f, 0.f, 0.f};
    for (int k0 = 0; k0 < K; k0 += 4) {
        v2f a, b;
        a.x = Arow[k0];
        a.y = Arow[k0 + 1];
        b.x = Bcol[(size_t)k0 * ldb];
        b.y = Bcol[(size_t)(k0 + 1) * ldb];
        acc = __builtin_amdgcn_wmma_f32_16x16x4_f32(false, a, false, b,
                                                    (short)0, acc, false, false);
    }
    float bv = bias ? bias[col] : 0.f;
#pragma unroll
    for (int v = 0; v < 8; ++v) {
        int row = rt * 16 + v + 8 * half;
        float val = acc[v] + bv;
        if (relu) val = fmaxf(val, 0.f);
        C[(size_t)row * ldc + col] = val;
    }
}

// ---------------- GCN layer pieces ----------------
// out (slice of xc, row stride 97): out[i*97+f] = h[i*F+f]*dis[i]^2 + bias[f]
__global__ void gcn_init(const float* __restrict__ h, int F, float* __restrict__ out,
                         const float* __restrict__ dis, const float* __restrict__ bias) {
    int i = blockIdx.x * blockDim.x + threadIdx.x;
    if (i >= NN * F) return;
    int node = i / F, f = i - node * F;
    float sn = dis[node]; sn *= sn;
    out[(size_t)node * XCF + f] = h[(size_t)node * F + f] * sn + bias[f];
}
// edge scatter, F=32: one lane per (edge, feature)
__global__ void gcn_edge32(const float* __restrict__ h, float* __restrict__ out,
                           const float* __restrict__ dis,
                           const int* __restrict__ src, const int* __restrict__ dst) {
    unsigned idx = blockIdx.x * blockDim.x + threadIdx.x;
    unsigned e = idx >> 5;
    int f = idx & 31;
    if (e >= NE) return;
    int s = src[e], d = dst[e];
    if (s == d) return;
    float en = dis[s] * dis[d];
    atomicAdd(&out[(size_t)d * XCF + f], h[(size_t)s * 32 + f] * en);
}
// edge scatter, F=1
__global__ void gcn_edge1(const float* __restrict__ h, float* __restrict__ out,
                          const float* __restrict__ dis,
                          const int* __restrict__ src, const int* __restrict__ dst) {
    unsigned e = blockIdx.x * blockDim.x + threadIdx.x;
    if (e >= NE) return;
    int s = src[e], d = dst[e];
    if (s == d) return;
    atomicAdd(&out[(size_t)d * XCF], h[s] * dis[s] * dis[d]);
}
__global__ void tanh_slice(float* __restrict__ out, int F) {
    int i = blockIdx.x * blockDim.x + threadIdx.x;
    if (i >= NN * F) return;
    int node = i / F, f = i - node * F;
    float* p = &out[(size_t)node * XCF + f];
    *p = tanhf(*p);
}
// layer-4 matmul: h[i] = dot(xc[i, 64:96], w4)
__global__ void mm32x1(const float* __restrict__ xcin, const float* __restrict__ w,
                       float* __restrict__ hout) {
    int i = blockIdx.x * blockDim.x + threadIdx.x;
    if (i >= NN) return;
    float acc = 0.f;
#pragma unroll
    for (int k = 0; k < 32; ++k) acc += xcin[(size_t)i * XCF + k] * w[k];
    hout[i] = acc;
}

// ---------------- SortPool ----------------
__global__ void count_nodes(const int* __restrict__ batch, int* __restrict__ counts) {
    int i = blockIdx.x * blockDim.x + threadIdx.x;
    if (i < NN) atomicAdd(&counts[batch[i]], 1);
}
__global__ void scan_counts(const int* __restrict__ counts, int* __restrict__ starts) {
    if (blockIdx.x == 0 && threadIdx.x == 0) {
        int run = 0;
        for (int g = 0; g < NG; ++g) { starts[g] = run; run += counts[g]; }
    }
}
// One wave per graph: stable top-K selection by (key desc, node-index asc),
// matching jnp.lexsort((-key, batch)) on already-batch-sorted nodes.
__global__ void sortpool(const float* __restrict__ xc, const int* __restrict__ counts,
                         const int* __restrict__ starts, float* __restrict__ pooled) {
    int g = blockIdx.x;
    int lane = threadIdx.x;          // blockDim 32
    int st = starts[g], cnt = counts[g];
    int rmax = cnt < KTOP ? cnt : KTOP;
    float prevK = 3.402823466e38f;   // sentinel: everything ranks after it
    int prevI = -1;
    for (int r = 0; r < KTOP; ++r) {
        if (r < rmax) {
            float bestK = -3.402823466e38f;
            int bestI = 0x7fffffff;
            for (int j = lane; j < cnt; j += 32) {
                int node = st + j;
                float key = xc[(size_t)node * XCF + 96];
                bool after = (key < prevK) || (key == prevK && node > prevI);
                if (after && (key > bestK || (key == bestK && node < bestI))) {
                    bestK = key; bestI = node;
                }
            }
            for (int off = 16; off; off >>= 1) {
                float ok = __shfl_xor(bestK, off, 32);
                int   oi = __shfl_xor(bestI, off, 32);
                if (ok > bestK || (ok == bestK && oi < bestI)) { bestK = ok; bestI = oi; }
            }
            prevK = bestK; prevI = bestI;
            for (int f = lane; f < XCF; f += 32)
                pooled[((size_t)g * KTOP + r) * XCF + f] = xc[(size_t)bestI * XCF + f];
        } else {
            for (int f = lane; f < XCF; f += 32)
                pooled[((size_t)g * KTOP + r) * XCF + f] = 0.f;
        }
    }
}

// ---------------- 1D conv head ----------------
// conv5: kernel 97, stride 97 over the (K*97) sequence -> per-row dot. out (NG,16,30)
__global__ void conv5relu(const float* __restrict__ pooled, const float* __restrict__ w,
                          const float* __restrict__ b, float* __restrict__ out5) {
    int i = blockIdx.x * blockDim.x + threadIdx.x;
    if (i >= NG * 16 * 30) return;
    int t = i % 30, c = (i / 30) % 16, g = i / (16 * 30);
    const float* row = pooled + ((size_t)g * KTOP + t) * XCF;
    const float* wc = w + c * XCF;
    float acc = b[c];
    for (int j = 0; j < XCF; ++j) acc += row[j] * wc[j];
    out5[i] = fmaxf(acc, 0.f);
}
__global__ void maxpool2(const float* __restrict__ out5, float* __restrict__ pool5) {
    int i = blockIdx.x * blockDim.x + threadIdx.x;
    if (i >= NG * 16 * 15) return;
    int t = i % 15, gc = i / 15;
    pool5[i] = fmaxf(out5[(size_t)gc * 30 + 2 * t], out5[(size_t)gc * 30 + 2 * t + 1]);
}
// conv6: (NG,16,15) -> (NG,32,11), flattened feature idx = c*11+t
__global__ void conv6relu(const float* __restrict__ pool5, const float* __restrict__ w,
                          const float* __restrict__ b, float* __restrict__ out6) {
    int i = blockIdx.x * blockDim.x + threadIdx.x;
    if (i >= NG * 352) return;
    int t = i % 11, c = (i / 11) % 32, g = i / 352;
    float acc = b[c];
    for (int ic = 0; ic < 16; ++ic) {
        const float* pr = pool5 + ((size_t)g * 16 + ic) * 15 + t;
        const float* wr = w + ((size_t)c * 16 + ic) * 5;
#pragma unroll
        for (int k = 0; k < 5; ++k) acc += pr[k] * wr[k];
    }
    out6[i] = fmaxf(acc, 0.f);
}

// ---------------- fc2 + log_softmax (one wave per graph) ----------------
__global__ void fc2_lsm(const float* __restrict__ fc1o, const float* __restrict__ w,
                        const float* __restrict__ b, float* __restrict__ out) {
    int g = blockIdx.x;
    int lane = threadIdx.x;          // 32
    float logit = -3.402823466e38f;
    if (lane < NC) {
        float acc = b[lane];
        for (int k = 0; k < 128; ++k) acc += fc1o[(size_t)g * 128 + k] * w[k * NC + lane];
        logit = acc;
    }
    float m = logit;
    for (int off = 16; off; off >>= 1) m = fmaxf(m, __shfl_xor(m, off, 32));
    float e = (lane < NC) ? expf(logit - m) : 0.f;
    float s = e;
    for (int off = 16; off; off >>= 1) s += __shfl_xor(s, off, 32);
    if (lane < NC) out[(size_t)g * NC + lane] = logit - m - logf(s);
}

// ---------------- launch ----------------
extern "C" void kernel_launch(void* const* d_in, const int* in_sizes, int n_in,
                              void* d_out, int out_size, void* d_ws, size_t ws_size,
                              hipStream_t stream) {
    const float* x       = (const float*)d_in[0];
    const float* w1      = (const float*)d_in[1];
    const float* b1      = (const float*)d_in[2];
    const float* w2      = (const float*)d_in[3];
    const float* b2      = (const float*)d_in[4];
    const float* w3      = (const float*)d_in[5];
    const float* b3      = (const float*)d_in[6];
    const float* w4      = (const float*)d_in[7];
    const float* b4      = (const float*)d_in[8];
    const float* conv5_w = (const float*)d_in[9];
    const float* conv5_b = (const float*)d_in[10];
    const float* conv6_w = (const float*)d_in[11];
    const float* conv6_b = (const float*)d_in[12];
    const float* fc1_w   = (const float*)d_in[13];
    const float* fc1_b   = (const float*)d_in[14];
    const float* fc2_w   = (const float*)d_in[15];
    const float* fc2_b   = (const float*)d_in[16];
    const int*   esrc    = (const int*)d_in[17];
    const int*   edst    = (const int*)d_in[18];
    const int*   batch   = (const int*)d_in[19];

    // workspace carving (~69 MB total, floats)
    float* ws = (float*)d_ws;
    size_t o = 0;
    float* dis    = ws + o; o += NN;                     // deg -> dis (in place)
    float* htmp   = ws + o; o += (size_t)NN * 32;        // per-layer matmul result
    float* xc     = ws + o; o += (size_t)NN * XCF;       // concat [x1|x2|x3|x4]
    float* pooled = ws + o; o += (size_t)NG * KTOP * XCF;
    float* out5   = ws + o; o += (size_t)NG * 16 * 30;
    float* pool5  = ws + o; o += (size_t)NG * 16 * 15;
    float* out6   = ws + o; o += (size_t)NG * 352;
    float* fc1o   = ws + o; o += (size_t)NG * 128;
    int* counts   = (int*)(ws + o); o += NG;
    int* starts   = (int*)(ws + o); o += NG;

    const int B = 256;
    auto cdiv = [](long long a, long long b) { return (int)((a + b - 1) / b); };

    // degree / dis
    zero_f<<<cdiv(NN, B), B, 0, stream>>>(dis, NN);
    deg_edges<<<cdiv(NE, B), B, 0, stream>>>(esrc, edst, dis);
    make_dis<<<cdiv(NN, B), B, 0, stream>>>(dis);

    const int RT = NN / 16;          // 6250 row tiles, exact
    // ---- GCN layer 1: x(100000x128) @ w1(128x32) ----
    wmma_mm<<<cdiv((long long)RT * 2, 8), B, 0, stream>>>(x, 128, w1, 32, htmp, 32,
                                                          128, RT, 2, nullptr, 0);
    gcn_init<<<cdiv((long long)NN * 32, B), B, 0, stream>>>(htmp, 32, xc + 0, dis, b1);
    gcn_edge32<<<cdiv((long long)NE * 32, B), B, 0, stream>>>(htmp, xc + 0, dis, esrc, edst);
    tanh_slice<<<cdiv((long long)NN * 32, B), B, 0, stream>>>(xc + 0, 32);
    // ---- layer 2: x1 @ w2 ----
    wmma_mm<<<cdiv((long long)RT * 2, 8), B, 0, stream>>>(xc + 0, XCF, w2, 32, htmp, 32,
                                                          32, RT, 2, nullptr, 0);
    gcn_init<<<cdiv((long long)NN * 32, B), B, 0, stream>>>(htmp, 32, xc + 32, dis, b2);
    gcn_edge32<<<cdiv((long long)NE * 32, B), B, 0, stream>>>(htmp, xc + 32, dis, esrc, edst);
    tanh_slice<<<cdiv((long long)NN * 32, B), B, 0, stream>>>(xc + 32, 32);
    // ---- layer 3: x2 @ w3 ----
    wmma_mm<<<cdiv((long long)RT * 2, 8), B, 0, stream>>>(xc + 32, XCF, w3, 32, htmp, 32,
                                                          32, RT, 2, nullptr, 0);
    gcn_init<<<cdiv((long long)NN * 32, B), B, 0, stream>>>(htmp, 32, xc + 64, dis, b3);
    gcn_edge32<<<cdiv((long long)NE * 32, B), B, 0, stream>>>(htmp, xc + 64, dis, esrc, edst);
    tanh_slice<<<cdiv((long long)NN * 32, B), B, 0, stream>>>(xc + 64, 32);
    // ---- layer 4: x3 @ w4 (32->1) ----
    mm32x1<<<cdiv(NN, B), B, 0, stream>>>(xc + 64, w4, htmp);
    gcn_init<<<cdiv(NN, B), B, 0, stream>>>(htmp, 1, xc + 96, dis, b4);
    gcn_edge1<<<cdiv(NE, B), B, 0, stream>>>(htmp, xc + 96, dis, esrc, edst);
    tanh_slice<<<cdiv(NN, B), B, 0, stream>>>(xc + 96, 1);

    // ---- SortPool ----
    zero_i<<<cdiv(NG, B), B, 0, stream>>>(counts, NG);
    count_nodes<<<cdiv(NN, B), B, 0, stream>>>(batch, counts);
    scan_counts<<<1, 1, 0, stream>>>(counts, starts);
    sortpool<<<NG, 32, 0, stream>>>(xc, counts, starts, pooled);

    // ---- conv head ----
    conv5relu<<<cdiv((long long)NG * 16 * 30, B), B, 0, stream>>>(pooled, conv5_w, conv5_b, out5);
    maxpool2<<<cdiv((long long)NG * 16 * 15, B), B, 0, stream>>>(out5, pool5);
    conv6relu<<<cdiv((long long)NG * 352, B), B, 0, stream>>>(pool5, conv6_w, conv6_b, out6);

    // ---- fc1 (1024x352 @ 352x128, bias+relu) via WMMA ----
    wmma_mm<<<cdiv((long long)(NG / 16) * 8, 8), B, 0, stream>>>(out6, 352, fc1_w, 128,
                                                                 fc1o, 128, 352,
                                                                 NG / 16, 8, fc1_b, 1);
    // ---- fc2 + log_softmax ----
    fc2_lsm<<<NG, 32, 0, stream>>>(fc1o, fc2_w, fc2_b, (float*)d_out);
}